// KgAdapterSentRGAT_71442486002191
// MI455X (gfx1250) — compile-verified
//
#include <hip/hip_runtime.h>
#include <math.h>

#define NN 100000
#define EE 1000000
#define DD 128
#define HH 4
#define DHH 32
#define NEG_SLOPE 0.2f

typedef float v2f __attribute__((ext_vector_type(2)));
typedef float v8f __attribute__((ext_vector_type(8)));

// ---------------------------------------------------------------------------
// float atomic max via sign-magnitude int trick (works for mixed signs, -inf)
// ---------------------------------------------------------------------------
__device__ __forceinline__ void atomicMaxFloat(float* addr, float val) {
    if (val >= 0.0f) {
        atomicMax((int*)addr, __float_as_int(val));
    } else {
        atomicMin((unsigned int*)addr, __float_as_uint(val));
    }
}

// ---------------------------------------------------------------------------
// init: m = -inf, z = 0
// ---------------------------------------------------------------------------
__global__ void k_init_nodes(float* __restrict__ m, float* __restrict__ z) {
    int i = blockIdx.x * blockDim.x + threadIdx.x;
    if (i < NN * HH) { m[i] = -INFINITY; z[i] = 0.0f; }
}

__global__ void k_zero(float* __restrict__ p, int n) {
    int i = blockIdx.x * blockDim.x + threadIdx.x;
    if (i < n) p[i] = 0.0f;
}

// ---------------------------------------------------------------------------
// h = x @ W  via V_WMMA_F32_16X16X4_F32 (fp32-exact matrix path)
// grid.x = N/16 row tiles, 8 waves/block = 8 column tiles of 16
// A 16x4: lane(0-15)=M, VGPR v + half(lane>=16) -> K = 2*half + v
// B 4x16: VGPR v + half -> K = 2*half + v, lane%16 = N
// C/D 16x16: VGPR r + half -> M = r + 8*half, lane%16 = N
// ---------------------------------------------------------------------------
__global__ __launch_bounds__(256) void k_gemm_wmma(const float* __restrict__ x,
                                                   const float* __restrict__ W,
                                                   float* __restrict__ h) {
    const int wave = threadIdx.x >> 5;   // 0..7  -> column tile
    const int lane = threadIdx.x & 31;
    const int half = lane >> 4;          // 0/1
    const int lm   = lane & 15;
    const int rowTile = blockIdx.x;      // 0..6249
    const int row = rowTile * 16 + lm;   // A row (M)
    const int col = wave * 16 + lm;      // B/D col (N)

    v8f acc = {0.f, 0.f, 0.f, 0.f, 0.f, 0.f, 0.f, 0.f};

    const float* xrow = x + (size_t)row * DD + 2 * half;   // A: x[row][k0+2h+v]
    const float* wcol = W + (size_t)(2 * half) * DD + col; // B: W[k0+2h+v][col]

#pragma unroll
    for (int k0 = 0; k0 < DD; k0 += 4) {
        v2f a, b;
        a[0] = xrow[k0];
        a[1] = xrow[k0 + 1];
        b[0] = wcol[(size_t)k0 * DD];
        b[1] = wcol[(size_t)k0 * DD + DD];
        acc = __builtin_amdgcn_wmma_f32_16x16x4_f32(
            /*neg_a=*/false, a, /*neg_b=*/false, b,
            /*c_mod=*/(short)0, acc, /*reuse_a=*/false, /*reuse_b=*/false);
    }

    float* hout = h + (size_t)(rowTile * 16 + 8 * half) * DD + wave * 16 + lm;
#pragma unroll
    for (int r = 0; r < 8; ++r)
        hout[(size_t)r * DD] = acc[r];
}

// ---------------------------------------------------------------------------
// per-edge attention logits; warp per edge; lane covers 4 floats (float4)
// lanes [8h..8h+7] cover head h (8 lanes * 4 = 32 = Dh)
// ---------------------------------------------------------------------------
__global__ __launch_bounds__(256) void k_edge_logits(
        const float* __restrict__ h, const float* __restrict__ rel_emb,
        const float* __restrict__ att_src, const float* __restrict__ att_dst,
        const float* __restrict__ att_rel,
        const int* __restrict__ edge_index, const int* __restrict__ edge_type,
        float* __restrict__ logits, float* __restrict__ m) {
    const int gw = (blockIdx.x * blockDim.x + threadIdx.x) >> 5;
    if (gw >= EE) return;
    const int lane = threadIdx.x & 31;
    const int src = edge_index[gw];
    const int dst = edge_index[EE + gw];
    const int et  = edge_type[gw];
    const int o = lane * 4;

    const float4 hs = *(const float4*)(h + (size_t)src * DD + o);
    const float4 hd = *(const float4*)(h + (size_t)dst * DD + o);
    const float4 rr = *(const float4*)(rel_emb + (size_t)et * DD + o);
    const float4 as = *(const float4*)(att_src + o);   // [H*Dh] flat: head*32 + j
    const float4 ad = *(const float4*)(att_dst + o);
    const float4 ar = *(const float4*)(att_rel + o);

    float p = hs.x * as.x + hs.y * as.y + hs.z * as.z + hs.w * as.w
            + hd.x * ad.x + hd.y * ad.y + hd.z * ad.z + hd.w * ad.w
            + rr.x * ar.x + rr.y * ar.y + rr.z * ar.z + rr.w * ar.w;

#pragma unroll
    for (int off = 1; off < 8; off <<= 1)
        p += __shfl_xor(p, off, 32);

    if ((lane & 7) == 0) {
        const int head = lane >> 3;
        const float lg = (p > 0.0f) ? p : NEG_SLOPE * p;  // leaky_relu
        logits[(size_t)gw * HH + head] = lg;
        atomicMaxFloat(&m[(size_t)dst * HH + head], lg);
    }
}

// ---------------------------------------------------------------------------
// e = exp(logit - m[dst]); store e into alpha region; z[dst] += e
// ---------------------------------------------------------------------------
__global__ void k_edge_exp(const int* __restrict__ edge_index,
                           const float* __restrict__ logits,
                           const float* __restrict__ m,
                           float* __restrict__ alpha, float* __restrict__ z) {
    int idx = blockIdx.x * blockDim.x + threadIdx.x;
    if (idx >= EE * HH) return;
    const int e  = idx >> 2;        // HH == 4
    const int hd = idx & 3;
    const int dst = edge_index[EE + e];
    const float ev = expf(logits[idx] - m[(size_t)dst * HH + hd]);
    alpha[idx] = ev;
    atomicAdd(&z[(size_t)dst * HH + hd], ev);
}

// ---------------------------------------------------------------------------
// alpha = e / (z[dst] + 1e-16)   (in place)
// ---------------------------------------------------------------------------
__global__ void k_alpha_norm(const int* __restrict__ edge_index,
                             const float* __restrict__ z,
                             float* __restrict__ alpha) {
    int idx = blockIdx.x * blockDim.x + threadIdx.x;
    if (idx >= EE * HH) return;
    const int e  = idx >> 2;
    const int hd = idx & 3;
    const int dst = edge_index[EE + e];
    alpha[idx] = alpha[idx] / (z[(size_t)dst * HH + hd] + 1e-16f);
}

// ---------------------------------------------------------------------------
// out[dst] += (h[src] + rel) * alpha ; warp per edge, float atomic scatter
// ---------------------------------------------------------------------------
__global__ __launch_bounds__(256) void k_aggregate(
        const float* __restrict__ h, const float* __restrict__ rel_emb,
        const int* __restrict__ edge_index, const int* __restrict__ edge_type,
        const float* __restrict__ alpha, float* __restrict__ out) {
    const int gw = (blockIdx.x * blockDim.x + threadIdx.x) >> 5;
    if (gw >= EE) return;
    const int lane = threadIdx.x & 31;
    const int src = edge_index[gw];
    const int dst = edge_index[EE + gw];
    const int et  = edge_type[gw];
    const int head = lane >> 3;
    const float a = alpha[(size_t)gw * HH + head];
    const int o = lane * 4;

    const float4 hs = *(const float4*)(h + (size_t)src * DD + o);
    const float4 rr = *(const float4*)(rel_emb + (size_t)et * DD + o);
    float* ob = out + (size_t)dst * DD + o;
    atomicAdd(ob + 0, (hs.x + rr.x) * a);
    atomicAdd(ob + 1, (hs.y + rr.y) * a);
    atomicAdd(ob + 2, (hs.z + rr.z) * a);
    atomicAdd(ob + 3, (hs.w + rr.w) * a);
}

// ---------------------------------------------------------------------------
// gelu (tanh approximation, jax.nn.gelu default)
// ---------------------------------------------------------------------------
__global__ void k_gelu(float* __restrict__ out) {
    int i = blockIdx.x * blockDim.x + threadIdx.x;
    if (i >= NN * DD) return;
    const float v = out[i];
    const float c = 0.7978845608028654f * (v + 0.044715f * v * v * v);
    out[i] = 0.5f * v * (1.0f + tanhf(c));
}

// ---------------------------------------------------------------------------
extern "C" void kernel_launch(void* const* d_in, const int* in_sizes, int n_in,
                              void* d_out, int out_size, void* d_ws, size_t ws_size,
                              hipStream_t stream) {
    (void)in_sizes; (void)n_in; (void)out_size; (void)ws_size;
    const float* x       = (const float*)d_in[0];
    const float* W       = (const float*)d_in[1];
    const float* rel_emb = (const float*)d_in[2];
    const float* att_src = (const float*)d_in[3];
    const float* att_dst = (const float*)d_in[4];
    const float* att_rel = (const float*)d_in[5];
    const int*   edge_ix = (const int*)d_in[6];
    const int*   edge_ty = (const int*)d_in[7];

    float* out   = (float*)d_out;                 // [N, D]
    float* alpha = out + (size_t)NN * DD;         // [E, H]

    float* ws     = (float*)d_ws;
    float* h      = ws;                           // N*D   (51.2 MB)
    float* logits = h + (size_t)NN * DD;          // E*H   (16 MB)
    float* m      = logits + (size_t)EE * HH;     // N*H
    float* z      = m + (size_t)NN * HH;          // N*H

    // init
    k_init_nodes<<<(NN * HH + 255) / 256, 256, 0, stream>>>(m, z);
    k_zero<<<(NN * DD + 255) / 256, 256, 0, stream>>>(out, NN * DD);

    // h = x @ W  (WMMA f32)
    k_gemm_wmma<<<NN / 16, 256, 0, stream>>>(x, W, h);

    // attention logits + segment max
    k_edge_logits<<<EE / 8, 256, 0, stream>>>(h, rel_emb, att_src, att_dst,
                                              att_rel, edge_ix, edge_ty,
                                              logits, m);
    // exp + segment sum
    k_edge_exp<<<(EE * HH + 255) / 256, 256, 0, stream>>>(edge_ix, logits, m,
                                                          alpha, z);
    // normalize
    k_alpha_norm<<<(EE * HH + 255) / 256, 256, 0, stream>>>(edge_ix, z, alpha);

    // weighted scatter aggregation
    k_aggregate<<<EE / 8, 256, 0, stream>>>(h, rel_emb, edge_ix, edge_ty,
                                            alpha, out);
    // gelu
    k_gelu<<<(NN * DD + 255) / 256, 256, 0, stream>>>(out);
}